// Mutual_Attention_7799660609556
// MI455X (gfx1250) — compile-verified
//
#include <hip/hip_runtime.h>
#include <hip/hip_bf16.h>
#include <math.h>

#define DIM    192
#define HEADS  8
#define CH     24
#define BATCH  8
#define NSP    16384            // 128*128 pixels
#define KSEG   16               // K-split segments for Gram partials
#define KEXT   (NSP / KSEG)     // 1024 pixels per segment
#define NC     128              // LDS chunk columns
#define ROWS   260              // row stride: x at +0, x*y at +132; 260%64==4 -> conflict-free
#define XYOFF  132
#define NCP    132              // padded stride for x tile in out_kernel
#define NCHUNKS (KEXT / NC)     // 8
#define GSP    196              // padded stride for G in LDS (196 % 64 == 4)
#define TILES  12               // 192 / 16

typedef float v2f __attribute__((ext_vector_type(2)));
typedef float v8f __attribute__((ext_vector_type(8)));

// ---------------------------------------------------------------------------
// Kernel 1: per-(b,seg) partial Gram matrices via f32 WMMA.
//   role 0 blocks: C1 = (x*y)(x*y)^T and C2 = (x*y) x^T, plus row sums of x, x*y
//   role 1 blocks: C3 = x x^T
// 512 threads = 16 waves in a 4x4 grid; each wave owns a 3x3 block of tiles.
// LDS 200KB -> 1 WG/WGP -> 4 waves/SIMD; launch_bounds(512,4) => ~256 VGPR budget.
// ---------------------------------------------------------------------------
__global__ __launch_bounds__(512, 4)
void gram_kernel(const float* __restrict__ x, const float* __restrict__ y,
                 float* __restrict__ cp12, float* __restrict__ cp3,
                 float* __restrict__ sxp, float* __restrict__ sxyp) {
  __shared__ float chs[DIM * ROWS];   // row r: x[r][0..127] at r*260, (x*y)[r][0..127] at r*260+132

  const int nseg = BATCH * KSEG;
  const int role = (blockIdx.x >= nseg) ? 1 : 0;
  const int bs   = blockIdx.x - role * nseg;
  const int b    = bs / KSEG;
  const int seg  = bs % KSEG;
  const int tid  = threadIdx.x;
  const int w    = tid >> 5;
  const int lane = tid & 31;
  const int half = lane >> 4;
  const int l16  = lane & 15;

  const int i0 = (w >> 2) * 3;        // 4 i-blocks of 3 tiles
  const int j0 = (w & 3) * 3;         // 4 j-blocks of 3 tiles
  int ar[3], br[3];
#pragma unroll
  for (int ii = 0; ii < 3; ++ii) ar[ii] = ((i0 + ii) * 16 + l16) * ROWS;
#pragma unroll
  for (int jj = 0; jj < 3; ++jj) br[jj] = ((j0 + jj) * 16 + l16) * ROWS;

  v8f acc1[3][3] = {};   // role0: C1 tiles, role1: C3 tiles
  v8f acc2[3][3] = {};   // role0: C2 tiles
  float sxacc = 0.f, sxyacc = 0.f;

  const float* xb = x + (size_t)b * DIM * NSP + (size_t)seg * KEXT;
  const float* yb = y + (size_t)b * DIM * NSP + (size_t)seg * KEXT;

  for (int chk = 0; chk < NCHUNKS; ++chk) {
    const int n0 = chk * NC;
    __syncthreads();
    if (role == 0) {
      for (int v = tid; v < DIM * (NC / 4); v += 512) {
        int r = v / (NC / 4), c4 = (v % (NC / 4)) * 4;
        float4 xv = *(const float4*)(xb + (size_t)r * NSP + n0 + c4);
        float4 yv = *(const float4*)(yb + (size_t)r * NSP + n0 + c4);
        float4 xyv = make_float4(xv.x * yv.x, xv.y * yv.y, xv.z * yv.z, xv.w * yv.w);
        *(float4*)(&chs[r * ROWS + c4])         = xv;
        *(float4*)(&chs[r * ROWS + XYOFF + c4]) = xyv;
      }
    } else {
      for (int v = tid; v < DIM * (NC / 4); v += 512) {
        int r = v / (NC / 4), c4 = (v % (NC / 4)) * 4;
        *(float4*)(&chs[r * ROWS + c4]) = *(const float4*)(xb + (size_t)r * NSP + n0 + c4);
      }
    }
    __syncthreads();

    if (role == 0 && tid < DIM) {
      for (int c = 0; c < NC; ++c) {
        sxacc  += chs[tid * ROWS + c];
        sxyacc += chs[tid * ROWS + XYOFF + c];
      }
    }

    if (role == 0) {
#pragma unroll 2
      for (int k0 = 0; k0 < NC; k0 += 4) {
        const int kk = k0 + 2 * half;
        v2f a[3], b1[3], b2[3];
#pragma unroll
        for (int ii = 0; ii < 3; ++ii)
          a[ii] = *(const v2f*)(&chs[ar[ii] + kk + XYOFF]);          // x*y rows (A)
#pragma unroll
        for (int jj = 0; jj < 3; ++jj) {
          b1[jj] = *(const v2f*)(&chs[br[jj] + kk + XYOFF]);         // x*y rows (B for C1)
          b2[jj] = *(const v2f*)(&chs[br[jj] + kk]);                 // x rows   (B for C2)
        }
#pragma unroll
        for (int ii = 0; ii < 3; ++ii)
#pragma unroll
          for (int jj = 0; jj < 3; ++jj) {
            acc1[ii][jj] = __builtin_amdgcn_wmma_f32_16x16x4_f32(false, a[ii], false, b1[jj], (short)0, acc1[ii][jj], false, false);
            acc2[ii][jj] = __builtin_amdgcn_wmma_f32_16x16x4_f32(false, a[ii], false, b2[jj], (short)0, acc2[ii][jj], false, false);
          }
      }
    } else {
#pragma unroll 2
      for (int k0 = 0; k0 < NC; k0 += 4) {
        const int kk = k0 + 2 * half;
        v2f a[3], b2[3];
#pragma unroll
        for (int ii = 0; ii < 3; ++ii)
          a[ii] = *(const v2f*)(&chs[ar[ii] + kk]);                  // x rows (A)
#pragma unroll
        for (int jj = 0; jj < 3; ++jj)
          b2[jj] = *(const v2f*)(&chs[br[jj] + kk]);                 // x rows (B)
#pragma unroll
        for (int ii = 0; ii < 3; ++ii)
#pragma unroll
          for (int jj = 0; jj < 3; ++jj)
            acc1[ii][jj] = __builtin_amdgcn_wmma_f32_16x16x4_f32(false, a[ii], false, b2[jj], (short)0, acc1[ii][jj], false, false);
      }
    }
  }

  const size_t base12 = (size_t)(b * KSEG + seg) * 2 * DIM * DIM;
  const size_t base3  = (size_t)(b * KSEG + seg) * DIM * DIM;
#pragma unroll
  for (int ii = 0; ii < 3; ++ii)
#pragma unroll
    for (int jj = 0; jj < 3; ++jj)
#pragma unroll
      for (int r = 0; r < 8; ++r) {
        int row = (i0 + ii) * 16 + r + 8 * half;
        int col = (j0 + jj) * 16 + l16;
        if (role == 0) {
          cp12[base12 + (size_t)row * DIM + col]             = acc1[ii][jj][r];
          cp12[base12 + DIM * DIM + (size_t)row * DIM + col] = acc2[ii][jj][r];
        } else {
          cp3[base3 + (size_t)row * DIM + col] = acc1[ii][jj][r];
        }
      }
  if (role == 0 && tid < DIM) {
    sxp [(b * KSEG + seg) * DIM + tid] = sxacc;
    sxyp[(b * KSEG + seg) * DIM + tid] = sxyacc;
  }
}

// ---------------------------------------------------------------------------
// Kernel 2: reduce K-segment partials -> Cfull[b][3][192][192], sx, sxy
// ---------------------------------------------------------------------------
__global__ __launch_bounds__(256)
void reduce_kernel(const float* __restrict__ cp12, const float* __restrict__ cp3,
                   const float* __restrict__ sxp,  const float* __restrict__ sxyp,
                   float* __restrict__ cf, float* __restrict__ sx, float* __restrict__ sxy) {
  const int b = blockIdx.x / 3, mat = blockIdx.x % 3;
  for (int idx = threadIdx.x; idx < DIM * DIM; idx += 256) {
    float s = 0.f;
    for (int seg = 0; seg < KSEG; ++seg) {
      size_t base = (size_t)(b * KSEG + seg);
      s += (mat < 2) ? cp12[(base * 2 + mat) * DIM * DIM + idx]
                     : cp3 [ base * DIM * DIM + idx];
    }
    cf[((size_t)b * 3 + mat) * DIM * DIM + idx] = s;
  }
  if (mat == 0 && threadIdx.x < DIM) {
    float a = 0.f, c = 0.f;
    for (int seg = 0; seg < KSEG; ++seg) {
      a += sxp [(b * KSEG + seg) * DIM + threadIdx.x];
      c += sxyp[(b * KSEG + seg) * DIM + threadIdx.x];
    }
    sx [b * DIM + threadIdx.x] = a;
    sxy[b * DIM + threadIdx.x] = c;
  }
}

// ---------------------------------------------------------------------------
// Kernel 3: per-(b,h) attention: S_raw = Wq C2 Wk^T + bias terms, norms from
// C1/C3 quadratic forms, temperature scale, softmax -> attn[b,h,24,24]
// ---------------------------------------------------------------------------
__global__ __launch_bounds__(192)
void attn_kernel(const float* __restrict__ cf, const float* __restrict__ sx,
                 const float* __restrict__ sxy,
                 const float* __restrict__ Wq, const float* __restrict__ bq,
                 const float* __restrict__ Wk, const float* __restrict__ bk,
                 const float* __restrict__ temp, float* __restrict__ attn) {
  const int b = blockIdx.x / HEADS, h = blockIdx.x % HEADS;
  const float* C1 = cf + ((size_t)b * 3 + 0) * DIM * DIM;
  const float* C2 = cf + ((size_t)b * 3 + 1) * DIM * DIM;
  const float* C3 = cf + ((size_t)b * 3 + 2) * DIM * DIM;

  __shared__ float u[CH][DIM];
  __shared__ float red[DIM][4];
  __shared__ float nq[CH], nk[CH], dq[CH], dk[CH];
  __shared__ float S[CH][CH];
  const int e = threadIdx.x;

  for (int cp = 0; cp < CH; ++cp) {
    const int c = h * CH + cp;
    const float* wqr = Wq + (size_t)c * DIM;
    const float* wkr = Wk + (size_t)c * DIM;
    float aq = 0.f, a1 = 0.f, a3 = 0.f;
    for (int a = 0; a < DIM; ++a) {
      float wq = wqr[a], wk = wkr[a];
      aq += wq * C2[a * DIM + e];
      a1 += wq * C1[a * DIM + e];
      a3 += wk * C3[a * DIM + e];
    }
    u[cp][e] = aq;
    red[e][0] = a1 * wqr[e];
    red[e][1] = a3 * wkr[e];
    red[e][2] = wqr[e] * sxy[b * DIM + e];
    red[e][3] = wkr[e] * sx [b * DIM + e];
    __syncthreads();
    for (int s = 96; s >= 6; s >>= 1) {
      if (e < s) for (int q = 0; q < 4; ++q) red[e][q] += red[e + s][q];
      __syncthreads();
    }
    if (e == 0) {
      float r0 = 0, r1 = 0, r2 = 0, r3 = 0;
      for (int i = 0; i < 6; ++i) { r0 += red[i][0]; r1 += red[i][1]; r2 += red[i][2]; r3 += red[i][3]; }
      nq[cp] = r0; nk[cp] = r1; dq[cp] = r2; dk[cp] = r3;
    }
    __syncthreads();
  }

  if (e < CH) {
    int c = h * CH + e;
    float nn = nq[e] + 2.f * bq[c] * dq[e] + (float)NSP * bq[c] * bq[c];
    nq[e] = fmaxf(sqrtf(fmaxf(nn, 0.f)), 1e-12f);
    nn = nk[e] + 2.f * bk[c] * dk[e] + (float)NSP * bk[c] * bk[c];
    nk[e] = fmaxf(sqrtf(fmaxf(nn, 0.f)), 1e-12f);
  }
  __syncthreads();

  for (int p = e; p < CH * CH; p += 192) {
    int cp = p / CH, dp = p % CH;
    int c = h * CH + cp, d = h * CH + dp;
    float s = (float)NSP * bq[c] * bk[d] + bk[d] * dq[cp] + bq[c] * dk[dp];
    const float* wkr = Wk + (size_t)d * DIM;
    for (int a = 0; a < DIM; ++a) s += u[cp][a] * wkr[a];
    S[cp][dp] = s / (nq[cp] * nk[dp]);
  }
  __syncthreads();

  if (e < CH) {
    const float t = temp[h];
    float mx = -1e30f;
#pragma unroll
    for (int d = 0; d < CH; ++d) { float z = S[e][d] * t; mx = fmaxf(mx, z); }
    float ex[CH], sum = 0.f;
#pragma unroll
    for (int d = 0; d < CH; ++d) { ex[d] = __expf(S[e][d] * t - mx); sum += ex[d]; }
    const float inv = 1.f / sum;
#pragma unroll
    for (int d = 0; d < CH; ++d)
      attn[(((size_t)(b * HEADS + h)) * CH + e) * CH + d] = ex[d] * inv;
  }
}

// ---------------------------------------------------------------------------
// Kernel 4: fold output side: M = Wo * blockdiag(attn);  G = M*Wv;  g = M*bv+bo
// ---------------------------------------------------------------------------
__global__ __launch_bounds__(256)
void fuse_kernel(const float* __restrict__ attn, const float* __restrict__ Wo,
                 const float* __restrict__ Wv, const float* __restrict__ bv,
                 const float* __restrict__ bo, float* __restrict__ G,
                 float* __restrict__ g) {
  const int b = blockIdx.x;
  __shared__ float M[DIM * DIM];
  __shared__ float A[HEADS * CH * CH];
  const int tid = threadIdx.x;
  for (int i = tid; i < HEADS * CH * CH; i += 256) A[i] = attn[(size_t)b * HEADS * CH * CH + i];
  __syncthreads();
  for (int p = tid; p < DIM * DIM; p += 256) {
    int o = p / DIM, j = p % DIM, h = j / CH, dp = j % CH;
    float s = 0.f;
    for (int cp = 0; cp < CH; ++cp) s += Wo[(size_t)o * DIM + h * CH + cp] * A[(h * CH + cp) * CH + dp];
    M[p] = s;
  }
  __syncthreads();
  for (int p = tid; p < DIM * DIM; p += 256) {
    int o = p / DIM, i = p % DIM;
    float s = 0.f;
    for (int j = 0; j < DIM; ++j) s += M[o * DIM + j] * Wv[(size_t)j * DIM + i];
    G[(size_t)b * DIM * DIM + p] = s;
  }
  if (tid < DIM) {
    float s = bo[tid];
    for (int j = 0; j < DIM; ++j) s += M[tid * DIM + j] * bv[j];
    g[b * DIM + tid] = s;
  }
}

// ---------------------------------------------------------------------------
// Kernel 5: final = G[b] @ x[b] + g[b]  (192x192 x 192x16384), WMMA f32.
// G and the x tile are LDS-resident; 8 waves x 12 acc tiles.
// ---------------------------------------------------------------------------
__global__ __launch_bounds__(256, 2)
void out_kernel(const float* __restrict__ x, const float* __restrict__ G,
                const float* __restrict__ g, float* __restrict__ out) {
  const int b  = blockIdx.x / (NSP / NC);
  const int nt = blockIdx.x % (NSP / NC);
  const int nbase = nt * NC;
  __shared__ float Gs[DIM * GSP];
  __shared__ float xt[DIM * NCP];
  __shared__ float gs[DIM];
  const int tid = threadIdx.x;

  for (int p = tid; p < DIM * (DIM / 4); p += 256) {
    int r = p / (DIM / 4), c4 = (p % (DIM / 4)) * 4;
    *(float4*)&Gs[r * GSP + c4] = *(const float4*)(G + (size_t)b * DIM * DIM + r * DIM + c4);
  }
  if (tid < DIM) gs[tid] = g[b * DIM + tid];
  const float* xb = x + (size_t)b * DIM * NSP + nbase;
  for (int p = tid; p < DIM * (NC / 4); p += 256) {
    int r = p / (NC / 4), c4 = (p % (NC / 4)) * 4;
    *(float4*)&xt[r * NCP + c4] = *(const float4*)(xb + (size_t)r * NSP + c4);
  }
  __syncthreads();

  const int w = tid >> 5, lane = tid & 31, half = lane >> 4, l16 = lane & 15;
  const int n0 = w * 16;
  int grow[TILES];
#pragma unroll
  for (int i = 0; i < TILES; ++i) grow[i] = (i * 16 + l16) * GSP;

  v8f acc[TILES] = {};
#pragma unroll 2
  for (int k0 = 0; k0 < DIM; k0 += 4) {
    const int kk = k0 + 2 * half;
    v2f bb;
    bb.x = xt[(kk)     * NCP + n0 + l16];
    bb.y = xt[(kk + 1) * NCP + n0 + l16];
#pragma unroll
    for (int i = 0; i < TILES; ++i) {
      v2f a = *(const v2f*)&Gs[grow[i] + kk];
      acc[i] = __builtin_amdgcn_wmma_f32_16x16x4_f32(false, a, false, bb, (short)0, acc[i], false, false);
    }
  }
#pragma unroll
  for (int i = 0; i < TILES; ++i) {
#pragma unroll
    for (int r = 0; r < 8; ++r) {
      int row = i * 16 + r + 8 * half;
      out[((size_t)b * DIM + row) * NSP + nbase + n0 + l16] = acc[i][r] + gs[row];
    }
  }
}

// ---------------------------------------------------------------------------
extern "C" void kernel_launch(void* const* d_in, const int* in_sizes, int n_in,
                              void* d_out, int out_size, void* d_ws, size_t ws_size,
                              hipStream_t stream) {
  (void)in_sizes; (void)n_in; (void)out_size; (void)ws_size;
  const float* x    = (const float*)d_in[0];
  const float* y    = (const float*)d_in[1];
  const float* Wq   = (const float*)d_in[2];
  const float* bq   = (const float*)d_in[3];
  const float* Wk   = (const float*)d_in[4];
  const float* bk   = (const float*)d_in[5];
  const float* Wv   = (const float*)d_in[6];
  const float* bv   = (const float*)d_in[7];
  const float* Wo   = (const float*)d_in[8];
  const float* bo   = (const float*)d_in[9];
  const float* temp = (const float*)d_in[10];
  float* out = (float*)d_out;
  float* ws  = (float*)d_ws;

  size_t o = 0;
  float* CP12 = ws + o; o += (size_t)BATCH * KSEG * 2 * DIM * DIM; // 37.7 MB
  float* CP3  = ws + o; o += (size_t)BATCH * KSEG * DIM * DIM;     // 18.9 MB
  float* SXP  = ws + o; o += (size_t)BATCH * KSEG * DIM;
  float* SXYP = ws + o; o += (size_t)BATCH * KSEG * DIM;
  float* CF   = ws + o; o += (size_t)BATCH * 3 * DIM * DIM;
  float* SX   = ws + o; o += (size_t)BATCH * DIM;
  float* SXY  = ws + o; o += (size_t)BATCH * DIM;
  float* ATT  = ws + o; o += (size_t)BATCH * HEADS * CH * CH;
  float* G    = ws + o; o += (size_t)BATCH * DIM * DIM;
  float* GB   = ws + o; o += (size_t)BATCH * DIM;

  gram_kernel  <<<2 * BATCH * KSEG,   512, 0, stream>>>(x, y, CP12, CP3, SXP, SXYP);
  reduce_kernel<<<BATCH * 3,          256, 0, stream>>>(CP12, CP3, SXP, SXYP, CF, SX, SXY);
  attn_kernel  <<<BATCH * HEADS,      192, 0, stream>>>(CF, SX, SXY, Wq, bq, Wk, bk, temp, ATT);
  fuse_kernel  <<<BATCH,              256, 0, stream>>>(ATT, Wo, Wv, bv, bo, G, GB);
  out_kernel   <<<BATCH * (NSP / NC), 256, 0, stream>>>(x, G, GB, out);
}